// HybridQLSTM_65481071396321
// MI455X (gfx1250) — compile-verified
//
#include <hip/hip_runtime.h>
#include <hip/hip_bf16.h>

// HybridQLSTM on MI455X (gfx1250, wave32).
//
// Phase 1: ZX[t,b,n] = x[t,b,:D] . Wl_n[:D]  via V_WMMA_F32_16X16X4_F32
//          (GEMM [T*B,512] x [512, 4 padded to 16]) -- streams x at full HBM BW.
// Phase 2: per-batch-row persistent scan (no cross-block deps): hx/cx in
//          registers, hx.Wlh reduced with wave32 shuffles + LDS partials,
//          ys written with non-temporal stores (stream > L2 capacity).

#define TT 512
#define BB 256
#define DD 512
#define HH 512

typedef __attribute__((ext_vector_type(2))) float v2f;
typedef __attribute__((ext_vector_type(8))) float v8f;

struct GateW {
  const float* Wl; const float* bl;
  const float* Wq; const float* bq;
  const float* Wm; const float* bm;
};
struct Params { GateW g[4]; };

// ---------------------------------------------------------------------------
// Phase 1: ZX precompute with f32 WMMA (16x16x4).
// Grid: 1024 blocks x 256 threads (8 waves); wave handles one 16-row tile.
// ---------------------------------------------------------------------------
__global__ __launch_bounds__(256) void qlstm_zx_wmma(
    const float* __restrict__ x,
    const float* __restrict__ Wl0, const float* __restrict__ Wl1,
    const float* __restrict__ Wl2, const float* __restrict__ Wl3,
    float* __restrict__ zx) {
  // B-operand staged as a 16x512 tile: rows 0..3 = gate weights, rows 4..15 = 0
  // (zero-padding instead of lane predication keeps EXEC all-1s around WMMA).
  __shared__ float wl_lds[16 * DD];
  const float* wls[4] = {Wl0, Wl1, Wl2, Wl3};
  for (int i = threadIdx.x; i < 16 * DD; i += 256) {
    int n = i >> 9;            // row (gate) 0..15
    int k = i & (DD - 1);      // k 0..511
    wl_lds[i] = (n < 4) ? wls[n][k] : 0.0f;
  }
  __syncthreads();

  const int lane = threadIdx.x & 31;
  const int wave = threadIdx.x >> 5;
  const int half = lane >> 4;      // 0: K={0,1}, 1: K={2,3} of each 4-chunk
  const int m    = lane & 15;      // A-matrix row within tile / B-matrix col
  const long tile = (long)blockIdx.x * 8 + wave;          // 8192 tiles of 16 rows
  const float* rowp = x + (tile * 16 + m) * (long)DD;     // A row for this lane

  v8f c = {};
#pragma unroll 8
  for (int k0 = 0; k0 < DD; k0 += 4) {
    const int kA = k0 + 2 * half;
    // A 16x4 f32: lane m, K = kA, kA+1
    v2f a = *(const v2f*)(rowp + kA);
    // B 4x16 f32 (padded): col m, same K pair; rows >=4 are zeros in LDS
    v2f b = *(const v2f*)(&wl_lds[m * DD + kA]);
    c = __builtin_amdgcn_wmma_f32_16x16x4_f32(
        /*neg_a=*/false, a, /*neg_b=*/false, b,
        /*c_mod=*/(short)0, c, /*reuse_a=*/false, /*reuse_b=*/false);
  }

  // C 16x16 f32: VGPR j holds row M=j (lanes 0-15) / M=j+8 (lanes 16-31), col=m.
  // Only cols 0..3 are real gates.
  if (m < 4) {
    const long rbase = tile * 16 + half * 8;
#pragma unroll
    for (int j = 0; j < 8; ++j) {
      zx[(rbase + j) * 4 + m] = c[j];
    }
  }
}

// ---------------------------------------------------------------------------
// Phase 2: sequential scan, one block per batch row (rows are independent).
// hx/cx live in registers (2 h-elements per thread). One barrier per step
// (double-buffered cross-wave partials).
// ---------------------------------------------------------------------------
__global__ __launch_bounds__(256) void qlstm_scan(
    const float* __restrict__ x, const float* __restrict__ zx,
    Params p, float* __restrict__ out) {
  __shared__ float wlh[4][HH];   // hidden-part of Wl (rows D..D+H-1)
  __shared__ float wmx[4][HH];   // Wm (rank-1 map weights)
  __shared__ float bmv[4][HH];   // bm
  __shared__ float wlx[4][DD];   // input-part of Wl (fallback path only)
  __shared__ float part[2][8][4];

  const int tid = threadIdx.x;
  const int b = blockIdx.x;
  const bool use_zx = (zx != nullptr);

  for (int i = tid; i < 4 * HH; i += 256) {
    const int n = i >> 9, h = i & (HH - 1);
    wlh[n][h] = p.g[n].Wl[DD + h];
    wmx[n][h] = p.g[n].Wm[h];
    bmv[n][h] = p.g[n].bm[h];
    if (!use_zx) wlx[n][h] = p.g[n].Wl[h];
  }
  float wq[4], bqv[4], blv[4];
#pragma unroll
  for (int n = 0; n < 4; ++n) {
    wq[n]  = p.g[n].Wq[0];
    bqv[n] = p.g[n].bq[0];
    blv[n] = p.g[n].bl[0];
  }
  __syncthreads();

  const int lane = tid & 31, wave = tid >> 5;
  const int h0 = tid, h1 = tid + 256;
  float hx0 = 0.f, hx1 = 0.f, cx0 = 0.f, cx1 = 0.f;

  for (int t = 0; t < TT; ++t) {
    // 4-gate partial dot: hx . Wl_n[D:]
    float pr[4];
#pragma unroll
    for (int n = 0; n < 4; ++n)
      pr[n] = hx0 * wlh[n][h0] + hx1 * wlh[n][h1];

    if (!use_zx) {  // fused fallback: fold in x . Wl_n[:D]
      const float* xr = x + ((long)t * BB + b) * (long)DD;
      const float x0 = xr[h0], x1 = xr[h1];
#pragma unroll
      for (int n = 0; n < 4; ++n)
        pr[n] += x0 * wlx[n][h0] + x1 * wlx[n][h1];
    }

    // wave32 butterfly reduce
#pragma unroll
    for (int off = 16; off >= 1; off >>= 1) {
#pragma unroll
      for (int n = 0; n < 4; ++n)
        pr[n] += __shfl_xor(pr[n], off, 32);
    }
    if (lane == 0) {
#pragma unroll
      for (int n = 0; n < 4; ++n) part[t & 1][wave][n] = pr[n];
    }
    __syncthreads();  // double-buffered partials -> one barrier per step

    // z_n -> q_n = tanh(z*Wq + bq)   (redundant per-thread scalar math)
    float q[4];
#pragma unroll
    for (int n = 0; n < 4; ++n) {
      float z = blv[n];
#pragma unroll
      for (int w = 0; w < 8; ++w) z += part[t & 1][w][n];
      if (use_zx) z += zx[((long)t * BB + b) * 4 + n];  // uniform -> s_load
      q[n] = tanhf(z * wq[n] + bqv[n]);
    }

    // gate construction (rank-1) + LSTM cell update, 2 h per thread
    {
      const float f = 1.f / (1.f + __expf(-(q[0] * wmx[0][h0] + bmv[0][h0])));
      const float i = 1.f / (1.f + __expf(-(q[1] * wmx[1][h0] + bmv[1][h0])));
      const float g = tanhf(q[2] * wmx[2][h0] + bmv[2][h0]);
      const float o = 1.f / (1.f + __expf(-(q[3] * wmx[3][h0] + bmv[3][h0])));
      cx0 = f * cx0 + i * g;
      hx0 = o * tanhf(cx0);
    }
    {
      const float f = 1.f / (1.f + __expf(-(q[0] * wmx[0][h1] + bmv[0][h1])));
      const float i = 1.f / (1.f + __expf(-(q[1] * wmx[1][h1] + bmv[1][h1])));
      const float g = tanhf(q[2] * wmx[2][h1] + bmv[2][h1]);
      const float o = 1.f / (1.f + __expf(-(q[3] * wmx[3][h1] + bmv[3][h1])));
      cx1 = f * cx1 + i * g;
      hx1 = o * tanhf(cx1);
    }

    // streaming write of ys[t,b,:] — non-temporal (268 MB > 192 MB L2)
    float* yrow = out + ((long)t * BB + b) * (long)HH;
    __builtin_nontemporal_store(hx0, yrow + h0);
    __builtin_nontemporal_store(hx1, yrow + h1);
  }

  // final hx, cx
  const long ysz = (long)TT * BB * HH;
  out[ysz + (long)b * HH + h0] = hx0;
  out[ysz + (long)b * HH + h1] = hx1;
  out[ysz + (long)BB * HH + (long)b * HH + h0] = cx0;
  out[ysz + (long)BB * HH + (long)b * HH + h1] = cx1;
}

extern "C" void kernel_launch(void* const* d_in, const int* in_sizes, int n_in,
                              void* d_out, int out_size, void* d_ws, size_t ws_size,
                              hipStream_t stream) {
  (void)in_sizes; (void)n_in; (void)out_size;
  const float* x = (const float*)d_in[0];
  Params p;
  for (int g = 0; g < 4; ++g) {
    const int base = 1 + 6 * g;                 // Wl, bl, Wq, bq, Wm, bm per gate
    p.g[g].Wl = (const float*)d_in[base + 0];
    p.g[g].bl = (const float*)d_in[base + 1];
    p.g[g].Wq = (const float*)d_in[base + 2];
    p.g[g].bq = (const float*)d_in[base + 3];
    p.g[g].Wm = (const float*)d_in[base + 4];
    p.g[g].bm = (const float*)d_in[base + 5];
  }

  float* zx = nullptr;
  const size_t zx_bytes = (size_t)TT * BB * 4 * sizeof(float);  // 2 MB
  if (ws_size >= zx_bytes) zx = (float*)d_ws;

  if (zx) {
    // (T*B)/(8 waves * 16 rows) = 1024 blocks
    qlstm_zx_wmma<<<1024, 256, 0, stream>>>(
        x, p.g[0].Wl, p.g[1].Wl, p.g[2].Wl, p.g[3].Wl, zx);
  }
  qlstm_scan<<<BB, 256, 0, stream>>>(x, zx, p, (float*)d_out);
}